// LSTMBlock_90202903150771
// MI455X (gfx1250) — compile-verified
//
#include <hip/hip_runtime.h>
#include <math.h>

#define B_   8
#define T_   4096
#define H_   512
#define G_   2048     // 4*H
#define NWG  16       // workgroups in recurrence (one per WGP)
#define LSTRIDE 516   // padded LDS row stride in floats (516 % 64 == 4 -> conflict-free)

typedef __attribute__((ext_vector_type(2))) float v2f;
typedef __attribute__((ext_vector_type(8))) float v8f;

// ---------------------------------------------------------------------------
// Kernel 1: LayerNorm over H=512. One block of 256 threads per row.
// ---------------------------------------------------------------------------
__global__ void ln_kernel(const float* __restrict__ x, const float* __restrict__ gamma,
                          const float* __restrict__ beta, float* __restrict__ res) {
    __shared__ float s1[256], s2[256];
    const int row = blockIdx.x;          // row = b*T + t (natural layout of x)
    const int tid = threadIdx.x;
    const float2 v = *(const float2*)(x + (size_t)row * H_ + tid * 2);
    s1[tid] = v.x + v.y;
    s2[tid] = v.x * v.x + v.y * v.y;
    __syncthreads();
    for (int off = 128; off > 0; off >>= 1) {
        if (tid < off) { s1[tid] += s1[tid + off]; s2[tid] += s2[tid + off]; }
        __syncthreads();
    }
    const float mu  = s1[0] * (1.0f / H_);
    const float var = s2[0] * (1.0f / H_) - mu * mu;
    const float rs  = rsqrtf(var + 1e-5f);
    const int n = tid * 2;
    float2 o;
    o.x = (v.x - mu) * rs * gamma[n]     + beta[n];
    o.y = (v.y - mu) * rs * gamma[n + 1] + beta[n + 1];
    *(float2*)(res + (size_t)row * H_ + n) = o;
}

// ---------------------------------------------------------------------------
// Dual-accumulator fp32 WMMA K-reduction over K=512 from two LDS rows.
// Two independent accumulator chains hide the WMMA->WMMA accumulate delay.
// ---------------------------------------------------------------------------
static __device__ __forceinline__ v8f wmma_k512(const float* __restrict__ Arow,
                                                const float* __restrict__ Brow,
                                                int kh) {
    v8f acc0 = {}, acc1 = {};
    for (int k0 = 0; k0 < H_; k0 += 8) {
        const v2f a0 = *(const v2f*)(Arow + k0 + 2 * kh);
        const v2f b0 = *(const v2f*)(Brow + k0 + 2 * kh);
        const v2f a1 = *(const v2f*)(Arow + k0 + 4 + 2 * kh);
        const v2f b1 = *(const v2f*)(Brow + k0 + 4 + 2 * kh);
        acc0 = __builtin_amdgcn_wmma_f32_16x16x4_f32(false, a0, false, b0,
                                                     (short)0, acc0, false, false);
        acc1 = __builtin_amdgcn_wmma_f32_16x16x4_f32(false, a1, false, b1,
                                                     (short)0, acc1, false, false);
    }
    return acc0 + acc1;
}

// ---------------------------------------------------------------------------
// Kernel 2: x_proj[t*8+b][g] = res[b][t][:] . W_ih[g][:] + b_ih[g] + b_hh[g]
// Block tile 32(M) x 64(N), full K=512 staged in LDS, 8 waves, one 16x16
// V_WMMA_F32_16X16X4_F32 tile per wave.
// ---------------------------------------------------------------------------
__global__ void xproj_kernel(const float* __restrict__ res, const float* __restrict__ W_ih,
                             const float* __restrict__ b_ih, const float* __restrict__ b_hh,
                             float* __restrict__ xp) {
    extern __shared__ float smem[];
    float* Alds = smem;                  // 32 x LSTRIDE
    float* Blds = smem + 32 * LSTRIDE;   // 64 x LSTRIDE
    const int tid = threadIdx.x;
    const int m0 = blockIdx.x * 32;      // M tile base; GEMM row r = t*8 + b
    const int n0 = blockIdx.y * 64;      // gate-column tile base

    for (int idx = tid; idx < 32 * 128; idx += 256) {       // A: 32 rows x 128 float4
        const int m = idx >> 7, kq = idx & 127;
        const int r = m0 + m;
        const int t = r >> 3, b = r & 7;
        *(float4*)(Alds + m * LSTRIDE + kq * 4) =
            *(const float4*)(res + ((size_t)(b * T_ + t)) * H_ + kq * 4);
    }
    for (int idx = tid; idx < 64 * 128; idx += 256) {       // B: W_ih rows g = n0+n
        const int n = idx >> 7, kq = idx & 127;
        *(float4*)(Blds + n * LSTRIDE + kq * 4) =
            *(const float4*)(W_ih + ((size_t)(n0 + n)) * H_ + kq * 4);
    }
    __syncthreads();

    const int lane = tid & 31, w = tid >> 5;
    const int wm = w >> 2, wn = w & 3;                      // 2x4 wave grid
    const int nl = lane & 15, kh = lane >> 4;
    const float* Arow = Alds + (wm * 16 + nl) * LSTRIDE;    // A: lane holds row M=nl
    const float* Brow = Blds + (wn * 16 + nl) * LSTRIDE;    // B: lane holds col N=nl
    const v8f acc = wmma_k512(Arow, Brow, kh);
    for (int i = 0; i < 8; ++i) {                           // C: VGPR i -> M=i (+8 for hi half)
        const int r = m0 + wm * 16 + i + kh * 8;
        const int g = n0 + wn * 16 + nl;
        xp[(size_t)r * G_ + g] = acc[i] + b_ih[g] + b_hh[g];
    }
}

// ---------------------------------------------------------------------------
// Kernel 3: the sequential LSTM recurrence, distributed over NWG=16 WGPs.
// WGP `wg` owns hidden columns j in [wg*32, wg*32+32): gate columns
// {j, 512+j, 1024+j, 1536+j}. Its 128x512 fp32 W_hh slice lives in LDS.
// Per step: stage h(t-1) (16x512, rows 8..15 zero), hoist xp[t] loads,
// 8 waves x (2x64) WMMA, epilogue (one (b,j) per thread, c in a register),
// then cluster/atomic split-barrier across the 16 WGPs.
// ---------------------------------------------------------------------------
__global__ void lstm_rec_kernel(const float* __restrict__ xp, const float* __restrict__ W_hh,
                                float* __restrict__ h_hist, unsigned* __restrict__ counter,
                                float* __restrict__ out /* full d_out; tail gets hT,cT */) {
    extern __shared__ float smem[];
    float* Wlds = smem;                       // 128 x LSTRIDE (264 KB)
    float* hlds = smem + 128 * LSTRIDE;       // 16 x LSTRIDE  (33 KB)
    float* glds = smem + 144 * LSTRIDE;       // 8 x 132 gate staging
    const int tid = threadIdx.x;
    const int wg  = blockIdx.x;               // 0..15

    // Load this WGP's W_hh slice: local col c -> W_hh row (c>>5)*512 + wg*32 + (c&31)
    for (int idx = tid; idx < 128 * 128; idx += 256) {
        const int c = idx >> 7, kq = idx & 127;
        const int grow = (c >> 5) * H_ + wg * 32 + (c & 31);
        *(float4*)(Wlds + c * LSTRIDE + kq * 4) =
            *(const float4*)(W_hh + (size_t)grow * H_ + kq * 4);
    }
    // Zero the padded h rows 8..15 once (never overwritten).
    for (int idx = tid; idx < 8 * 128; idx += 256) {
        const int m = 8 + (idx >> 7), kq = idx & 127;
        *(float4*)(hlds + m * LSTRIDE + kq * 4) = make_float4(0.f, 0.f, 0.f, 0.f);
    }

    const int lane = tid & 31, w = tid >> 5;  // 8 waves, wave w -> N tile w
    const int nl = lane & 15, kh = lane >> 4;
    const int eb = tid >> 5, ej = tid & 31;   // epilogue: thread owns (b=eb, j=ej)
    const int jg = wg * 32 + ej;              // global hidden column
    float creg = 0.0f;                        // cell state c[b][jg]
    const unsigned nwgu = (unsigned)gridDim.x;
    const size_t OUT_TAIL = (size_t)B_ * T_ * H_;
    const float* Arow = hlds + nl * LSTRIDE;
    const float* Brow = Wlds + (w * 16 + nl) * LSTRIDE;

    for (int t = 0; t < T_; ++t) {
        // Stage h(t-1) rows 0..7 (zeros at t==0).
        for (int idx = tid; idx < 8 * 128; idx += 256) {
            const int b = idx >> 7, kq = idx & 127;
            float4 hv = make_float4(0.f, 0.f, 0.f, 0.f);
            if (t > 0)
                hv = *(const float4*)(h_hist + ((size_t)(t - 1) * B_ + b) * H_ + kq * 4);
            *(float4*)(hlds + b * LSTRIDE + kq * 4) = hv;
        }
        // Hoist xp[t] gate-bias loads: they only depend on t, so issue them now
        // and let them fly under the WMMA loop.
        const size_t xpb = ((size_t)t * B_ + eb) * G_;
        const float xi = xp[xpb +         jg];
        const float xf = xp[xpb +  512 + jg];
        const float xg = xp[xpb + 1024 + jg];
        const float xo = xp[xpb + 1536 + jg];
        __syncthreads();

        const v8f acc = wmma_k512(Arow, Brow, kh);
        if (kh == 0) {                        // lanes 0..15 hold rows M=0..7 in acc[0..7]
            for (int i = 0; i < 8; ++i)
                glds[i * 132 + w * 16 + nl] = acc[i];
        }
        __syncthreads();

        // Epilogue: gates = glds + x_proj; torch order i,f,g,o.
        const float gi = glds[eb * 132 +       ej] + xi;
        const float gf = glds[eb * 132 +  32 + ej] + xf;
        const float gg = glds[eb * 132 +  64 + ej] + xg;
        const float go = glds[eb * 132 +  96 + ej] + xo;
        const float iv = 1.0f / (1.0f + expf(-gi));
        const float fv = 1.0f / (1.0f + expf(-gf));
        const float gv = tanhf(gg);
        const float ov = 1.0f / (1.0f + expf(-go));
        creg = fv * creg + iv * gv;
        const float hval = ov * tanhf(creg);
        h_hist[((size_t)t * B_ + eb) * H_ + jg] = hval;
        if (t == T_ - 1) {
            out[OUT_TAIL + (size_t)eb * H_ + jg] = hval;                       // hT
            out[OUT_TAIL + (size_t)B_ * H_ + (size_t)eb * H_ + jg] = creg;     // cT
        }

        // Cross-WGP split-barrier for this step.
        __threadfence();
        __syncthreads();
        if (tid == 0) {
            __builtin_amdgcn_s_cluster_barrier();   // NOP when not clustered
            __hip_atomic_fetch_add(counter, 1u, __ATOMIC_ACQ_REL, __HIP_MEMORY_SCOPE_AGENT);
            const unsigned target = (unsigned)(t + 1) * nwgu;
            while (__hip_atomic_load(counter, __ATOMIC_ACQUIRE, __HIP_MEMORY_SCOPE_AGENT) < target)
                __builtin_amdgcn_s_sleep(1);
        }
        __syncthreads();
    }
}

// ---------------------------------------------------------------------------
// Kernel 4: gate = silu(res @ W_lin^T + b_lin); out = h_hist*gate + res.
// Same GEMM structure as kernel 2; residual comes straight from the A tile.
// ---------------------------------------------------------------------------
__global__ void gate_fuse_kernel(const float* __restrict__ res, const float* __restrict__ W_lin,
                                 const float* __restrict__ b_lin, const float* __restrict__ h_hist,
                                 float* __restrict__ out) {
    extern __shared__ float smem[];
    float* Alds = smem;                  // 32 x LSTRIDE
    float* Blds = smem + 32 * LSTRIDE;   // 64 x LSTRIDE
    const int tid = threadIdx.x;
    const int m0 = blockIdx.x * 32;      // rows r = b*T + t (natural res layout)
    const int n0 = blockIdx.y * 64;

    for (int idx = tid; idx < 32 * 128; idx += 256) {
        const int m = idx >> 7, kq = idx & 127;
        *(float4*)(Alds + m * LSTRIDE + kq * 4) =
            *(const float4*)(res + (size_t)(m0 + m) * H_ + kq * 4);
    }
    for (int idx = tid; idx < 64 * 128; idx += 256) {
        const int n = idx >> 7, kq = idx & 127;
        *(float4*)(Blds + n * LSTRIDE + kq * 4) =
            *(const float4*)(W_lin + (size_t)(n0 + n) * H_ + kq * 4);
    }
    __syncthreads();

    const int lane = tid & 31, w = tid >> 5;
    const int wm = w >> 2, wn = w & 3;
    const int nl = lane & 15, kh = lane >> 4;
    const float* Arow = Alds + (wm * 16 + nl) * LSTRIDE;
    const float* Brow = Blds + (wn * 16 + nl) * LSTRIDE;
    const v8f acc = wmma_k512(Arow, Brow, kh);
    for (int i = 0; i < 8; ++i) {
        const int m = wm * 16 + i + kh * 8;      // row within block tile
        const int r = m0 + m;
        const int n = n0 + wn * 16 + nl;
        const float val  = acc[i] + b_lin[n];
        const float gate = val / (1.0f + expf(-val));   // silu
        const int b = r >> 12, t = r & (T_ - 1);
        const float hv  = h_hist[((size_t)t * B_ + b) * H_ + n];
        const float rres = Alds[m * LSTRIDE + n];       // res[r][n] (n < 512 = K)
        out[(size_t)r * H_ + n] = hv * gate + rres;
    }
}

// ---------------------------------------------------------------------------
extern "C" void kernel_launch(void* const* d_in, const int* in_sizes, int n_in,
                              void* d_out, int out_size, void* d_ws, size_t ws_size,
                              hipStream_t stream) {
    (void)in_sizes; (void)n_in; (void)out_size; (void)ws_size;
    const float* x     = (const float*)d_in[0];
    const float* W_ih  = (const float*)d_in[1];
    const float* W_hh  = (const float*)d_in[2];
    const float* b_ih  = (const float*)d_in[3];
    const float* b_hh  = (const float*)d_in[4];
    const float* gamma = (const float*)d_in[5];
    const float* beta  = (const float*)d_in[6];
    const float* W_lin = (const float*)d_in[7];
    const float* b_lin = (const float*)d_in[8];
    float* out = (float*)d_out;

    char* ws = (char*)d_ws;
    float*    res    = (float*)(ws);                                     // 64 MB
    float*    xp     = (float*)(ws + (size_t)64 * 1024 * 1024);          // 256 MB [T][B][4H]
    float*    h_hist = (float*)(ws + (size_t)320 * 1024 * 1024);         // 64 MB  [T][B][H]
    unsigned* cnt    = (unsigned*)(ws + (size_t)384 * 1024 * 1024);      // barrier counter

    const size_t gemm_lds = (size_t)(32 + 64) * LSTRIDE * sizeof(float);       // 198144 B
    const size_t rec_lds  = (size_t)(144 * LSTRIDE + 8 * 132) * sizeof(float); // 301440 B

    // Allow >64KB dynamic LDS on CDNA5 (320KB/WGP). Host-side attribute set;
    // deterministic and not a stream operation (graph-capture safe).
    (void)hipFuncSetAttribute((const void*)xproj_kernel,
                              hipFuncAttributeMaxDynamicSharedMemorySize, (int)gemm_lds);
    (void)hipFuncSetAttribute((const void*)lstm_rec_kernel,
                              hipFuncAttributeMaxDynamicSharedMemorySize, (int)rec_lds);
    (void)hipFuncSetAttribute((const void*)gate_fuse_kernel,
                              hipFuncAttributeMaxDynamicSharedMemorySize, (int)gemm_lds);

    ln_kernel<<<B_ * T_, 256, 0, stream>>>(x, gamma, beta, res);
    xproj_kernel<<<dim3(1024, 32), 256, gemm_lds, stream>>>(res, W_ih, b_ih, b_hh, xp);
    hipMemsetAsync(cnt, 0, sizeof(unsigned), stream);
    lstm_rec_kernel<<<NWG, 256, rec_lds, stream>>>(xp, W_hh, h_hist, cnt, out);
    gate_fuse_kernel<<<dim3(1024, 8), 256, gemm_lds, stream>>>(res, W_lin, b_lin, h_hist, out);
}